// projector_30416958390643
// MI455X (gfx1250) — compile-verified
//
#include <hip/hip_runtime.h>
#include <hip/hip_bf16.h>

// Problem constants (fixed by the reference).
#define BB 16
#define CC 3
#define HH 384
#define WW 1280
#define EPSF 1e-7f

typedef float v2f __attribute__((ext_vector_type(2)));
typedef float v8f __attribute__((ext_vector_type(8)));

// ---------------------------------------------------------------------------
// Kernel 1: per-batch composition  M = K * R(axis-angle) * inv(K),  T = K * t
// One wave (32 lanes) per batch. The 3x3 product M = (K*R) * invK is done with
// V_WMMA_F32_16X16X4_F32, padding the 3x3s into the 16x4 (A) / 4x16 (B)
// fragment layouts from ISA 7.12.2:
//   A: VGPR0 = {lanes 0-15: (M=lane,K=0), lanes 16-31: (M=lane-16,K=2)}
//      VGPR1 = {K=1, K=3}
//   B: VGPR0 = {lanes 0-15: (K=0,N=lane), lanes 16-31: (K=2,N=lane-16)}
//      VGPR1 = {K=1, K=3}
//   D: acc[i] = row i, column = lane (lanes 0-15)
// No divergence before the WMMA (EXEC must be all ones).
// ---------------------------------------------------------------------------
__global__ __launch_bounds__(32) void compose_30416958390643(
    const float* __restrict__ pose, const float* __restrict__ intr,
    float* __restrict__ params) {
  const int b = blockIdx.x;
  const int lane = threadIdx.x;
  const float* K = intr + b * 9;
  const float* P = pose + b * 6;

  // axis-angle -> R (all lanes redundantly; loads are wave-uniform)
  float ax = P[0], ay = P[1], az = P[2];
  float theta = sqrtf(ax * ax + ay * ay + az * az);
  float it = 1.0f / (theta + EPSF);
  float x = ax * it, y = ay * it, z = az * it;
  float cth = cosf(theta), sth = sinf(theta), t1 = 1.0f - cth;
  float R[9];
  R[0] = t1 * x * x + cth; R[1] = t1 * x * y - sth * z; R[2] = t1 * z * x + sth * y;
  R[3] = t1 * x * y + sth * z; R[4] = t1 * y * y + cth; R[5] = t1 * y * z - sth * x;
  R[6] = t1 * z * x - sth * y; R[7] = t1 * y * z + sth * x; R[8] = t1 * z * z + cth;

  float k0 = K[0], k1 = K[1], k2 = K[2], k3 = K[3], k4 = K[4],
        k5 = K[5], k6 = K[6], k7 = K[7], k8 = K[8];

  // inv(K) via adjugate
  float det = k0 * (k4 * k8 - k5 * k7) - k1 * (k3 * k8 - k5 * k6) +
              k2 * (k3 * k7 - k4 * k6);
  float id = 1.0f / det;
  float iK[9];
  iK[0] = (k4 * k8 - k5 * k7) * id; iK[1] = (k2 * k7 - k1 * k8) * id; iK[2] = (k1 * k5 - k2 * k4) * id;
  iK[3] = (k5 * k6 - k3 * k8) * id; iK[4] = (k0 * k8 - k2 * k6) * id; iK[5] = (k2 * k3 - k0 * k5) * id;
  iK[6] = (k3 * k7 - k4 * k6) * id; iK[7] = (k1 * k6 - k0 * k7) * id; iK[8] = (k0 * k4 - k1 * k3) * id;

  // KR = K * R (27 FLOPs, scalar)
  float KR[9];
  #pragma unroll
  for (int i = 0; i < 3; ++i) {
    float a0 = K[i * 3 + 0], a1 = K[i * 3 + 1], a2 = K[i * 3 + 2];
    KR[i * 3 + 0] = a0 * R[0] + a1 * R[3] + a2 * R[6];
    KR[i * 3 + 1] = a0 * R[1] + a1 * R[4] + a2 * R[7];
    KR[i * 3 + 2] = a0 * R[2] + a1 * R[5] + a2 * R[8];
  }

  const int m = lane & 15;
  const int hi = lane >> 4;

#if __has_builtin(__builtin_amdgcn_wmma_f32_16x16x4_f32)
  // M = KR (16x4, rows 0-2 / K 0-2 live) x iK (4x16, K 0-2 / cols 0-2 live)
  v2f afrag, bfrag;
  afrag.x = (m < 3) ? KR[m * 3 + (hi ? 2 : 0)] : 0.0f;
  afrag.y = (hi || m >= 3) ? 0.0f : KR[m * 3 + 1];
  bfrag.x = (m < 3) ? iK[(hi ? 2 : 0) * 3 + m] : 0.0f;
  bfrag.y = (hi || m >= 3) ? 0.0f : iK[3 + m];
  v8f acc = {};
  acc = __builtin_amdgcn_wmma_f32_16x16x4_f32(
      /*neg_a=*/false, afrag, /*neg_b=*/false, bfrag,
      /*c_mod=*/(short)0, acc, /*reuse_a=*/false, /*reuse_b=*/false);
  if (!hi && m < 3) {
    params[b * 12 + 0 * 3 + m] = acc[0];  // M[0][m]
    params[b * 12 + 1 * 3 + m] = acc[1];  // M[1][m]
    params[b * 12 + 2 * 3 + m] = acc[2];  // M[2][m]
  }
#else
  // scalar fallback (keeps the file compiling if the builtin is absent)
  if (lane < 9) {
    int i = lane / 3, j = lane % 3;
    params[b * 12 + lane] =
        KR[i * 3 + 0] * iK[j] + KR[i * 3 + 1] * iK[3 + j] + KR[i * 3 + 2] * iK[6 + j];
  }
#endif

  if (lane == 0) {
    float t0 = P[3], t1v = P[4], t2 = P[5];
    params[b * 12 + 9]  = k0 * t0 + k1 * t1v + k2 * t2;
    params[b * 12 + 10] = k3 * t0 + k4 * t1v + k5 * t2;
    params[b * 12 + 11] = k6 * t0 + k7 * t1v + k8 * t2;
  }
}

// ---------------------------------------------------------------------------
// Kernel 2: fused warp + bilinear sample. One pixel per thread.
// coord = depth * (M . [x,y,1]) + T ; sample at (coord.x/z, coord.y/z).
// Depth reads and output writes are non-temporal (streamed once) so the
// 94 MB source image stays resident in the 192 MB L2 for the scattered
// gathers (depth is i.i.d. -> gather targets are effectively random).
// ---------------------------------------------------------------------------
__global__ __launch_bounds__(256) void warp_30416958390643(
    const float* __restrict__ src, const float* __restrict__ depth,
    const float* __restrict__ params, float* __restrict__ out) {
  int idx = blockIdx.x * 256 + threadIdx.x;
  const int total = BB * HH * WW;
  if (idx >= total) return;
  int w = idx % WW;
  int t = idx / WW;
  int h = t % HH;
  int b = t / HH;

  const float* Pp = params + b * 12;  // block-uniform
  float m00 = Pp[0], m01 = Pp[1], m02 = Pp[2];
  float m10 = Pp[3], m11 = Pp[4], m12 = Pp[5];
  float m20 = Pp[6], m21 = Pp[7], m22 = Pp[8];
  float T0 = Pp[9], T1 = Pp[10], T2 = Pp[11];

  float d = __builtin_nontemporal_load(depth + idx);
  float fx = (float)w, fy = (float)h;
  float c0 = fmaf(d, fmaf(m00, fx, fmaf(m01, fy, m02)), T0);
  float c1 = fmaf(d, fmaf(m10, fx, fmaf(m11, fy, m12)), T1);
  float c2 = fmaf(d, fmaf(m20, fx, fmaf(m21, fy, m22)), T2);
  float z = c2 + EPSF;
  float invz = 1.0f / z;
  float px = c0 * invz;  // == denormalized sample x (grid norm/denorm cancel)
  float py = c1 * invz;

  float x0 = floorf(px), y0 = floorf(py);
  float x1 = x0 + 1.0f, y1 = y0 + 1.0f;
  float wx1 = px - x0, wx0 = 1.0f - wx1;
  float wy1 = py - y0, wy0 = 1.0f - wy1;

  const float Wm1 = (float)(WW - 1), Hm1 = (float)(HH - 1);
  bool inx0 = (x0 >= 0.0f) && (x0 <= Wm1);
  bool inx1 = (x1 >= 0.0f) && (x1 <= Wm1);
  bool iny0 = (y0 >= 0.0f) && (y0 <= Hm1);
  bool iny1 = (y1 >= 0.0f) && (y1 <= Hm1);
  int xi0 = (int)fminf(fmaxf(x0, 0.0f), Wm1);
  int xi1 = (int)fminf(fmaxf(x1, 0.0f), Wm1);
  int yi0 = (int)fminf(fmaxf(y0, 0.0f), Hm1);
  int yi1 = (int)fminf(fmaxf(y1, 0.0f), Hm1);

  float w00 = (inx0 && iny0) ? wx0 * wy0 : 0.0f;
  float w10 = (inx1 && iny0) ? wx1 * wy0 : 0.0f;
  float w01 = (inx0 && iny1) ? wx0 * wy1 : 0.0f;
  float w11 = (inx1 && iny1) ? wx1 * wy1 : 0.0f;

  const float* sb = src + (size_t)b * (CC * HH * WW);
  int i00 = yi0 * WW + xi0, i10 = yi0 * WW + xi1;
  int i01 = yi1 * WW + xi0, i11 = yi1 * WW + xi1;
  int obase = b * (CC * HH * WW) + h * WW + w;

  #pragma unroll
  for (int ch = 0; ch < CC; ++ch) {
    const float* sc = sb + ch * (HH * WW);
    float v = fmaf(w00, sc[i00],
              fmaf(w10, sc[i10],
              fmaf(w01, sc[i01],
                   w11 * sc[i11])));
    __builtin_nontemporal_store(v, out + obase + ch * (HH * WW));
  }
}

extern "C" void kernel_launch(void* const* d_in, const int* in_sizes, int n_in,
                              void* d_out, int out_size, void* d_ws, size_t ws_size,
                              hipStream_t stream) {
  const float* src   = (const float*)d_in[0];  // (B,C,H,W)
  const float* depth = (const float*)d_in[1];  // (B,1,H,W)
  const float* pose  = (const float*)d_in[2];  // (B,6)
  const float* intr  = (const float*)d_in[3];  // (B,3,3)
  float* out = (float*)d_out;
  float* params = (float*)d_ws;  // 16 batches x 12 floats = 768 B

  compose_30416958390643<<<dim3(BB), dim3(32), 0, stream>>>(pose, intr, params);

  const int total = BB * HH * WW;
  const int blocks = (total + 255) / 256;
  warp_30416958390643<<<dim3(blocks), dim3(256), 0, stream>>>(src, depth, params, out);
}